// GNNBlock_71176198029766
// MI455X (gfx1250) — compile-verified
//
#include <hip/hip_runtime.h>

typedef __attribute__((ext_vector_type(2))) float    v2f;
typedef __attribute__((ext_vector_type(8))) float    v8f;
typedef __attribute__((ext_vector_type(4))) unsigned v4u;
typedef __attribute__((ext_vector_type(8))) unsigned v8u;

#define D        128
#define NHEAD    4
#define DH       32
#define KNBR     8
#define TOPK     4
#define NEG_SLOPE 0.2f
#define ALPHA    0.05f

#define N1   20000
#define E1   (N1*KNBR)      // 160000
#define CHUNK 20000
#define NCHUNK (E1/CHUNK)   // 8

// ---------------------------------------------------------------------------
// Generic WMMA GEMM: Y[M,NCOLS] = act( sum_s gather(A_s)@W_s + bias + acc*Y )
// K fixed = 128 per source, W row-stride fixed = 128.
// Block = 256 threads (8 waves); each wave owns a 16-row strip; block = 128 rows.
// Weight chunks staged via TDM (tensor_load_to_lds, TENSORcnt);
// gathered A strips staged via global_load_async_to_lds_b128 (ASYNCcnt).
// ---------------------------------------------------------------------------
struct GemmSrc {
  const float* A;
  const int*   idx;
  int          idxmode;   // 0: row, 1: idx[row], 2: row>>3
  int          lda;
  const float* W;         // row-major, row stride 128
};

template<int NTILES, int NSRC>
__global__ __launch_bounds__(256)
void gemm_wmma(GemmSrc s0, GemmSrc s1, GemmSrc s2,
               const float* bias, float* __restrict__ Y, int ldy,
               int M, int accum, int dorelu)
{
  constexpr int NCOLS = NTILES * 16;
  __shared__ float sW[32 * NCOLS];     // K-chunk of weights [kk][nn]
  __shared__ float sA[8 * 16 * 32];    // per-wave A strip chunk [m][kk]

  const int  tid  = threadIdx.x;
  const int  wave = tid >> 5;
  const int  lane = tid & 31;
  const bool lo   = lane < 16;
  const int  hn   = lane & 15;
  const int  strip = blockIdx.x * 128 + wave * 16;
  float* sAw = sA + wave * (16 * 32);

  v8f acc[NTILES];
  #pragma unroll
  for (int t = 0; t < NTILES; ++t) {
    #pragma unroll
    for (int r = 0; r < 8; ++r) {
      int row = strip + (lo ? r : r + 8);
      float v = 0.f;
      if (accum && row < M) v = Y[(size_t)row * ldy + t * 16 + hn];
      acc[t][r] = v;
    }
  }

  #pragma unroll
  for (int s = 0; s < NSRC; ++s) {
    GemmSrc S = (s == 0) ? s0 : ((s == 1) ? s1 : s2);
    #pragma unroll 1
    for (int kc = 0; kc < 4; ++kc) {
      const int k0 = kc * 32;
      __syncthreads();   // protect LDS buffers from previous iteration's readers

      // ---- TDM: DMA the [32 x NCOLS] weight chunk into sW (wave 0 issues) ----
      if (wave == 0) {
        unsigned long long ga = (unsigned long long)(const void*)(S.W + (size_t)k0 * 128);
        unsigned ldsoff = (unsigned)(size_t)(void*)sW;  // low 32 bits = LDS offset
        v4u g0;
        g0[0] = 1u;                                   // count=1, user descriptor
        g0[1] = ldsoff;                               // lds_addr
        g0[2] = (unsigned)ga;                         // global_addr[31:0]
        g0[3] = ((unsigned)(ga >> 32) & 0x01FFFFFFu)  // global_addr[56:32]
                | (2u << 30);                         // type = 2 (image)
        v8u g1;
        g1[0] = 2u << 16;                             // data_size = 4 bytes
        g1[1] = 0u;                                   // tensor_dim0[15:0] = 0 (of 1<<20)
        g1[2] = 0x0010u;                              // tensor_dim0[31:16] ; tensor_dim1 lo
        g1[3] = 0x0010u | ((unsigned)NCOLS << 16);    // tensor_dim1[31:16] ; tile_dim0
        g1[4] = 32u;                                  // tile_dim1 = 32 rows, tile_dim2 = 0
        g1[5] = 128u;                                 // tensor_dim0_stride = 128 elements
        g1[6] = 0u;
        g1[7] = 0u;
        asm volatile("tensor_load_to_lds %0, %1" :: "s"(g0), "s"(g1) : "memory");
      }

      // ---- async copy: gathered A strip [16 x 32] -> sAw (per-lane b128) ----
      #pragma unroll
      for (int q = 0; q < 4; ++q) {
        int idx4 = lane + 32 * q;
        int m  = idx4 >> 3;
        int kk = (idx4 & 7) * 4;
        int row = strip + m;
        int rr  = (row < M) ? row : (M - 1);
        int srow = (S.idxmode == 0) ? rr : ((S.idxmode == 1) ? S.idx[rr] : (rr >> 3));
        const float* gp = S.A + (size_t)srow * S.lda + k0 + kk;
        unsigned ldsa = (unsigned)(size_t)(void*)(sAw + m * 32 + kk);
        asm volatile("global_load_async_to_lds_b128 %0, %1, off"
                     :: "v"(ldsa), "v"(gp) : "memory");
      }

      if (wave == 0) __builtin_amdgcn_s_wait_tensorcnt(0);
      asm volatile("s_wait_asynccnt 0x0" ::: "memory");
      __syncthreads();

      // ---- 8 K-steps of 4, per ISA f32 16x16x4 fragment layout ----
      #pragma unroll
      for (int kb = 0; kb < 8; ++kb) {
        const int ksel = lo ? 0 : 2;
        v2f a;
        a.x = sAw[hn * 32 + kb * 4 + ksel];
        a.y = sAw[hn * 32 + kb * 4 + ksel + 1];
        #pragma unroll
        for (int t = 0; t < NTILES; ++t) {
          v2f b;
          b.x = sW[(kb * 4 + ksel)     * NCOLS + t * 16 + hn];
          b.y = sW[(kb * 4 + ksel + 1) * NCOLS + t * 16 + hn];
          acc[t] = __builtin_amdgcn_wmma_f32_16x16x4_f32(
              false, a, false, b, (short)0, acc[t], false, false);
        }
      }
    }
  }

  #pragma unroll
  for (int t = 0; t < NTILES; ++t) {
    int col = t * 16 + hn;
    float bval = bias ? bias[col] : 0.f;
    #pragma unroll
    for (int r = 0; r < 8; ++r) {
      int row = strip + (lo ? r : r + 8);
      if (row < M) {
        float y = acc[t][r] + bval;
        if (dorelu) y = fmaxf(y, 0.f);
        Y[(size_t)row * ldy + col] = y;
      }
    }
  }
}

// ---------------------------------------------------------------------------
// aeproj[j,h] = sum_d W_rel[j, h*32+d] * a_e[h,d];  eebias[h] = b_rel[h*32:]·a_e[h]
// ---------------------------------------------------------------------------
__global__ void prep_kernel(const float* __restrict__ Wrel, const float* __restrict__ brel,
                            const float* __restrict__ ae,
                            float* __restrict__ aeproj, float* __restrict__ eebias)
{
  int j = threadIdx.x;  // 0..127
  #pragma unroll
  for (int h = 0; h < NHEAD; ++h) {
    float s = 0.f;
    for (int d = 0; d < DH; ++d) s += Wrel[(size_t)j * 128 + h * DH + d] * ae[h * DH + d];
    aeproj[j * NHEAD + h] = s;
  }
  if (j < NHEAD) {
    float s = 0.f;
    for (int d = 0; d < DH; ++d) s += brel[j * DH + d] * ae[j * DH + d];
    eebias[j] = s;
  }
}

// ---------------------------------------------------------------------------
// el/er: per (node,head) 32-dim dot with a_l / a_r
// ---------------------------------------------------------------------------
__global__ void eler_kernel(const float* __restrict__ feat, const float* __restrict__ al,
                            const float* __restrict__ ar,
                            float* __restrict__ el, float* __restrict__ er, int M)
{
  int i = blockIdx.x * blockDim.x + threadIdx.x;
  if (i >= M * NHEAD) return;
  int m = i >> 2, h = i & 3;
  const float* f = feat + (size_t)m * D + h * DH;
  float sl = 0.f, sr = 0.f;
  #pragma unroll
  for (int d = 0; d < DH; ++d) { sl += f[d] * al[h * DH + d]; sr += f[d] * ar[h * DH + d]; }
  el[i] = sl; er[i] = sr;
}

// ---------------------------------------------------------------------------
// Attention + aggregation: one wave per destination node (32 lanes x float4).
// Writes Y[gm,:] = aggfeat + ALPHA*feat[gm] + b_rel  and  z[ml,h,:] = sum_k a*rel.
// ---------------------------------------------------------------------------
__global__ __launch_bounds__(256)
void attn_kernel(const int* __restrict__ nbr, const float* __restrict__ feat,
                 const float* __restrict__ el, const float* __restrict__ er,
                 const float* __restrict__ rel, const float* __restrict__ aeproj,
                 const float* __restrict__ eebias, const float* __restrict__ brel,
                 float* __restrict__ z, float* __restrict__ Y,
                 int node_base, int nchunk)
{
  const int wave = threadIdx.x >> 5;
  const int lane = threadIdx.x & 31;
  const int ml   = blockIdx.x * 8 + wave;
  if (ml >= nchunk) return;
  const int gm = node_base + ml;
  const int c0 = lane * 4;
  const int hc = c0 >> 5;     // head owning this lane's columns

  int nb[KNBR];
  #pragma unroll
  for (int k = 0; k < KNBR; ++k) nb[k] = nbr[(size_t)ml * KNBR + k];

  float rv[KNBR][4];
  #pragma unroll
  for (int k = 0; k < KNBR; ++k) {
    float4 v = *(const float4*)(rel + ((size_t)ml * KNBR + k) * D + c0);
    rv[k][0] = v.x; rv[k][1] = v.y; rv[k][2] = v.z; rv[k][3] = v.w;
  }

  float ap[4][NHEAD];
  #pragma unroll
  for (int t = 0; t < 4; ++t)
    #pragma unroll
    for (int h = 0; h < NHEAD; ++h) ap[t][h] = aeproj[(c0 + t) * NHEAD + h];

  float eb[NHEAD], erv[NHEAD];
  #pragma unroll
  for (int h = 0; h < NHEAD; ++h) { eb[h] = eebias[h]; erv[h] = er[(size_t)gm * NHEAD + h]; }

  // e[k][h] = leaky( el[nbr] + er + ee ),  ee via projected rel + wave reduction
  float e[KNBR][NHEAD];
  #pragma unroll
  for (int k = 0; k < KNBR; ++k) {
    float p[NHEAD] = {0.f, 0.f, 0.f, 0.f};
    #pragma unroll
    for (int t = 0; t < 4; ++t)
      #pragma unroll
      for (int h = 0; h < NHEAD; ++h) p[h] += rv[k][t] * ap[t][h];
    #pragma unroll
    for (int h = 0; h < NHEAD; ++h) {
      float v = p[h];
      #pragma unroll
      for (int off = 16; off >= 1; off >>= 1) v += __shfl_xor(v, off, 32);
      p[h] = v;
    }
    #pragma unroll
    for (int h = 0; h < NHEAD; ++h) {
      float x = el[(size_t)nb[k] * NHEAD + h] + erv[h] + p[h] + eb[h];
      e[k][h] = (x >= 0.f) ? x : NEG_SLOPE * x;
    }
  }

  // per-head top-4 mask (ties kept, matching s >= kth) + softmax
  float aw[NHEAD][KNBR];
  #pragma unroll
  for (int h = 0; h < NHEAD; ++h) {
    bool used[KNBR];
    #pragma unroll
    for (int k = 0; k < KNBR; ++k) used[k] = false;
    float kth = 0.f, maxv = 0.f;
    #pragma unroll
    for (int t = 0; t < TOPK; ++t) {
      float bm = -3.4e38f; int bi = 0;
      #pragma unroll
      for (int k = 0; k < KNBR; ++k)
        if (!used[k] && e[k][h] > bm) { bm = e[k][h]; bi = k; }
      if (t == 0) maxv = bm;
      kth = bm;
      #pragma unroll
      for (int k = 0; k < KNBR; ++k) used[k] = used[k] || (k == bi);
    }
    float ssum = 0.f;
    #pragma unroll
    for (int k = 0; k < KNBR; ++k) {
      float x  = e[k][h];
      float ex = (x >= kth) ? __expf(x - maxv) : 0.f;
      aw[h][k] = ex; ssum += ex;
    }
    float inv = 1.f / ssum;
    #pragma unroll
    for (int k = 0; k < KNBR; ++k) aw[h][k] *= inv;
  }

  // weights for this lane's column-head (avoid dynamic register indexing)
  float ac[KNBR];
  #pragma unroll
  for (int k = 0; k < KNBR; ++k)
    ac[k] = (hc == 0) ? aw[0][k] : ((hc == 1) ? aw[1][k] : ((hc == 2) ? aw[2][k] : aw[3][k]));

  float o0, o1, o2, o3;
  {
    float4 f = *(const float4*)(feat + (size_t)gm * D + c0);
    o0 = ALPHA * f.x + brel[c0];
    o1 = ALPHA * f.y + brel[c0 + 1];
    o2 = ALPHA * f.z + brel[c0 + 2];
    o3 = ALPHA * f.w + brel[c0 + 3];
  }
  #pragma unroll
  for (int k = 0; k < KNBR; ++k) {
    float4 f = *(const float4*)(feat + (size_t)nb[k] * D + c0);
    o0 += ac[k] * f.x; o1 += ac[k] * f.y; o2 += ac[k] * f.z; o3 += ac[k] * f.w;
  }
  *(float4*)(Y + (size_t)gm * D + c0) = make_float4(o0, o1, o2, o3);

  #pragma unroll
  for (int h = 0; h < NHEAD; ++h) {
    float z0 = 0.f, z1 = 0.f, z2 = 0.f, z3 = 0.f;
    #pragma unroll
    for (int k = 0; k < KNBR; ++k) {
      z0 += aw[h][k] * rv[k][0]; z1 += aw[h][k] * rv[k][1];
      z2 += aw[h][k] * rv[k][2]; z3 += aw[h][k] * rv[k][3];
    }
    *(float4*)(z + (size_t)ml * (NHEAD * D) + h * D + c0) = make_float4(z0, z1, z2, z3);
  }
}

// ---------------------------------------------------------------------------
extern "C" void kernel_launch(void* const* d_in, const int* in_sizes, int n_in,
                              void* d_out_, int out_size, void* d_ws, size_t ws_size,
                              hipStream_t stream)
{
  (void)in_sizes; (void)n_in; (void)out_size; (void)ws_size;
  const int*   nbr1      = (const int*)d_in[0];
  const int*   nbr2      = (const int*)d_in[1];
  const float* node_feat = (const float*)d_in[2];
  const float* len_h     = (const float*)d_in[3];
  const float* angle_h   = (const float*)d_in[4];
  const float* W_ent1 = (const float*)d_in[5];
  const float* b_ent1 = (const float*)d_in[6];
  const float* W_rel1 = (const float*)d_in[7];
  const float* b_rel1 = (const float*)d_in[8];
  const float* a_l1   = (const float*)d_in[9];
  const float* a_r1   = (const float*)d_in[10];
  const float* a_e1   = (const float*)d_in[11];
  const float* W_ent2 = (const float*)d_in[12];
  const float* b_ent2 = (const float*)d_in[13];
  const float* W_rel2 = (const float*)d_in[14];
  const float* b_rel2 = (const float*)d_in[15];
  const float* a_l2   = (const float*)d_in[16];
  const float* a_r2   = (const float*)d_in[17];
  const float* a_e2   = (const float*)d_in[18];
  const float* W_t    = (const float*)d_in[19];
  const float* b_t    = (const float*)d_in[20];

  float* out = (float*)d_out_;
  float* ws  = (float*)d_ws;
  float* g1out = out;                      // [N1, 128]
  float* g2out = out + (size_t)N1 * D;     // [E1, 128]

  // workspace layout (floats); total ~37.3M floats (~150 MB)
  float* ae1   = ws;                                      // 512
  float* eeb1  = ws + 512;                                // 4
  float* ae2   = ws + 516;                                // 512
  float* eeb2  = ws + 1028;                               // 4
  float* feat1 = ws + 1040;                               // N1*D
  float* el1   = feat1 + (size_t)N1 * D;                  // N1*4
  float* er1   = el1 + (size_t)N1 * NHEAD;
  float* el2   = er1 + (size_t)N1 * NHEAD;                // E1*4
  float* er2   = el2 + (size_t)E1 * NHEAD;
  float* zbuf  = er2 + (size_t)E1 * NHEAD;                // CHUNK*4*D
  float* trip  = zbuf + (size_t)CHUNK * NHEAD * D;        // CHUNK*D
  float* feat2 = trip + (size_t)CHUNK * D;                // E1*D

  const int GB = (CHUNK + 127) / 128;   // 157 blocks per GEMM chunk
  GemmSrc zsrc{nullptr, nullptr, 0, 0, nullptr};

  // projections for the ee term
  prep_kernel<<<1, 128, 0, stream>>>(W_rel1, b_rel1, a_e1, ae1, eeb1);
  prep_kernel<<<1, 128, 0, stream>>>(W_rel2, b_rel2, a_e2, ae2, eeb2);

  // ---- GNN layer 1 ----
  {
    GemmSrc s{node_feat, nullptr, 0, D, W_ent1};
    gemm_wmma<8, 1><<<GB, 256, 0, stream>>>(s, zsrc, zsrc, b_ent1, feat1, D, N1, 0, 0);
  }
  eler_kernel<<<(N1 * NHEAD + 255) / 256, 256, 0, stream>>>(feat1, a_l1, a_r1, el1, er1, N1);
  attn_kernel<<<(N1 + 7) / 8, 256, 0, stream>>>(nbr1, feat1, el1, er1, len_h,
                                                ae1, eeb1, b_rel1, zbuf, g1out, 0, N1);
  for (int h = 0; h < NHEAD; ++h) {   // g1out[:, h*32:+32] += Z_h @ W_rel1_h, then relu
    GemmSrc s{zbuf + h * D, nullptr, 0, NHEAD * D, W_rel1 + h * DH};
    gemm_wmma<2, 1><<<GB, 256, 0, stream>>>(s, zsrc, zsrc, nullptr, g1out + h * DH, D, N1, 1, 1);
  }

  // ---- Triplet embedding + feat2, chunked ----
  for (int ch = 0; ch < NCHUNK; ++ch) {
    size_t base = (size_t)ch * CHUNK;
    GemmSrc sh{g1out, nbr1 + base, 1, D, W_t};                       // head_h = g1out[src]
    GemmSrc sl{len_h + base * D, nullptr, 0, D, W_t + 128 * 128};    // len_h
    GemmSrc se{g1out + (base >> 3) * D, nullptr, 2, D, W_t + 256 * 128}; // end_h = g1out[e>>3]
    gemm_wmma<8, 3><<<GB, 256, 0, stream>>>(sh, sl, se, b_t, trip, D, CHUNK, 0, 1);
    GemmSrc sf{trip, nullptr, 0, D, W_ent2};
    gemm_wmma<8, 1><<<GB, 256, 0, stream>>>(sf, zsrc, zsrc, b_ent2, feat2 + base * D, D, CHUNK, 0, 0);
  }

  // ---- GNN layer 2, chunked ----
  eler_kernel<<<(E1 * NHEAD + 255) / 256, 256, 0, stream>>>(feat2, a_l2, a_r2, el2, er2, E1);
  for (int ch = 0; ch < NCHUNK; ++ch) {
    size_t base = (size_t)ch * CHUNK;
    attn_kernel<<<CHUNK / 8, 256, 0, stream>>>(nbr2 + base * KNBR, feat2, el2, er2,
                                               angle_h + base * KNBR * D,
                                               ae2, eeb2, b_rel2, zbuf, g2out,
                                               (int)base, CHUNK);
    for (int h = 0; h < NHEAD; ++h) {
      GemmSrc s{zbuf + h * D, nullptr, 0, NHEAD * D, W_rel2 + h * DH};
      gemm_wmma<2, 1><<<GB, 256, 0, stream>>>(s, zsrc, zsrc, nullptr,
                                              g2out + base * D + h * DH, D, CHUNK, 1, 1);
    }
  }
}